// EdgeDegreeEmbedding_72782515798382
// MI455X (gfx1250) — compile-verified
//
#include <hip/hip_runtime.h>
#include <hip/hip_bf16.h>

typedef __attribute__((ext_vector_type(2))) float v2f;
typedef __attribute__((ext_vector_type(8))) float v8f;
typedef __attribute__((ext_vector_type(4))) unsigned int v4u;
typedef __attribute__((ext_vector_type(8))) int v8i;
typedef __attribute__((ext_vector_type(4))) int v4i;

#define FULLC 49          // (LMAX+1)^2
#define M0    7           // m = 0 coefficients
#define MALL  29          // reduced coefficient stride in wigner_inv_envelope
#define CCH   128         // sphere channels
#define HID   128         // hidden width
#define OUTD  896         // M0 * CCH
#define TE    16          // edges per workgroup (one WMMA M-tile)
#define XS    132         // padded LDS stride for xbuf/hbuf (bank-conflict free)
#define RADS  900         // padded LDS stride for rad buffer
#define INV_RESCALE (1.0f / 23.395f)

// LDS layout in floats:
//   xbuf   [TE*XS]      = 2112
//   hbuf   [TE*XS]      = 2112
//   radbuf [TE*RADS]    = 14400
//   wbuf   [TE*49*7]    = 5488   (filled by TDM tensor_load_to_lds)
//   lnred  [2*128]      = 256
//   lnmv   [TE*2]       = 32
#define LDS_FLOATS (2112 + 2112 + 14400 + 5488 + 256 + 32)

__device__ __forceinline__ void atom_add_f32(float* p, float v) {
    __hip_atomic_fetch_add(p, v, __ATOMIC_RELAXED, __HIP_MEMORY_SCOPE_AGENT);
}

__global__ void __launch_bounds__(256)
copy_f4_kernel(const float4* __restrict__ src, float4* __restrict__ dst, int n4) {
    int i = blockIdx.x * blockDim.x + threadIdx.x;
    if (i < n4) dst[i] = src[i];
}

__global__ void __launch_bounds__(128)
edge_deg_fused_kernel(const float* __restrict__ x_edge,
                      const float* __restrict__ wig,     // [E,49,29]
                      const float* __restrict__ W1,      // [128,128]
                      const float* __restrict__ b1,
                      const float* __restrict__ g1,
                      const float* __restrict__ be1,
                      const float* __restrict__ W2,      // [128,896]
                      const float* __restrict__ b2,
                      const int*   __restrict__ eidx,    // [2,E]
                      float*       __restrict__ out,     // [N,49,128]
                      int E) {
    extern __shared__ float lds[];
    float* xbuf   = lds;
    float* hbuf   = xbuf + TE * XS;
    float* radbuf = hbuf + TE * XS;
    float* wbuf   = radbuf + TE * RADS;
    float* lnred  = wbuf + TE * FULLC * M0;
    float* lnmv   = lnred + 2 * 128;

    const int tid   = threadIdx.x;
    const int lane  = tid & 31;
    const int wave  = tid >> 5;
    const int half  = lane >> 4;   // 0: lanes 0-15, 1: lanes 16-31
    const int lm    = lane & 15;
    const int edge0 = blockIdx.x * TE;
    int nE = E - edge0; if (nE > TE) nE = TE;

    // ---- TDM: DMA the used wigner columns [TE*49 rows][7 floats], row
    //      stride 29, straight into wbuf. One issue from wave 0; rows past
    //      nE*49 are OOB in the tensor -> hardware zero-fill.
    if (wave == 0) {
        const unsigned wlds = (unsigned)(size_t)wbuf;   // flat LDS addr: low 32 bits = LDS offset
        const unsigned long long ga =
            (unsigned long long)(size_t)(wig + (size_t)edge0 * FULLC * MALL);
        const unsigned rows = (unsigned)nE * FULLC;

        v4u g0;
        g0[0] = 1u;                                          // count=1, no gather
        g0[1] = wlds;                                        // lds_addr
        g0[2] = (unsigned)ga;                                // global_addr[31:0]
        g0[3] = (unsigned)((ga >> 32) & 0x1FFFFFFu)          // global_addr[56:32]
              | (2u << 30);                                  // type = 2 ("image")
        v8i g1;
        g1[0] = (int)(2u << 16);                             // data_size = 4B, wg_mask=0
        g1[1] = (int)((unsigned)MALL << 16);                 // tensor_dim0 = 29 (lo half)
        g1[2] = (int)(rows << 16);                           // dim0 hi=0 | tensor_dim1 lo = rows
        g1[3] = (int)((unsigned)M0 << 16);                   // dim1 hi=0 | tile_dim0 = 7
        g1[4] = (int)(TE * FULLC);                           // tile_dim1 = 784 | tile_dim2 = 0
        g1[5] = MALL;                                        // tensor_dim0_stride = 29 (lo 32)
        g1[6] = 0;                                           // stride hi | dim1_stride lo
        g1[7] = 0;
        v4i gz4 = {0, 0, 0, 0};
        v8i gz8 = {0, 0, 0, 0, 0, 0, 0, 0};
        __builtin_amdgcn_tensor_load_to_lds(g0, g1, gz4, gz4, gz8, 0);
    }

    // ---- stage x_edge tile into LDS (coalesced, overlaps with TDM) ----
    for (int i = tid; i < TE * HID; i += 128) {
        int e = i >> 7, k = i & 127;
        xbuf[e * XS + k] = (e < nE) ? x_edge[(size_t)(edge0 + e) * HID + k] : 0.0f;
    }
    if (wave == 0) __builtin_amdgcn_s_wait_tensorcnt(0);
    __syncthreads();

    // ---- GEMM1: h[16x128] = xtile @ W1, f32 WMMA 16x16x4 ----
    // A/B VGPR v: lanes0-15 K=k0+v, lanes16-31 K=k0+v+2.
    for (int nt = wave * 2; nt < wave * 2 + 2; ++nt) {
        const int n0 = nt * 16;
        v8f acc = {};
        for (int k0 = 0; k0 < HID; k0 += 4) {
            const int ka = k0 + half * 2;
            v2f a = *(const v2f*)(xbuf + lm * XS + ka);       // ds_load_b64
            v2f b;
            b[0] = W1[(size_t)ka * HID + n0 + lm];
            b[1] = W1[(size_t)(ka + 1) * HID + n0 + lm];
            acc = __builtin_amdgcn_wmma_f32_16x16x4_f32(false, a, false, b,
                                                        (short)0, acc, false, false);
        }
        const float bb = b1[n0 + lm];
        #pragma unroll
        for (int i = 0; i < 8; ++i)     // C VGPR i -> row i (+8 for upper half-wave)
            hbuf[(i + half * 8) * XS + n0 + lm] = acc[i] + bb;
    }
    __syncthreads();

    // ---- LayerNorm + SiLU over the 128 features of each edge ----
    {
        const int e = tid >> 3, seg = tid & 7;
        float s = 0.0f, sq = 0.0f;
        for (int k = seg * 16; k < seg * 16 + 16; ++k) {
            float v = hbuf[e * XS + k];
            s += v; sq += v * v;
        }
        lnred[e * 8 + seg]       = s;
        lnred[128 + e * 8 + seg] = sq;
    }
    __syncthreads();
    if (tid < TE) {
        float s = 0.0f, sq = 0.0f;
        for (int j = 0; j < 8; ++j) { s += lnred[tid * 8 + j]; sq += lnred[128 + tid * 8 + j]; }
        const float mu  = s * (1.0f / 128.0f);
        const float var = sq * (1.0f / 128.0f) - mu * mu;
        lnmv[tid * 2]     = mu;
        lnmv[tid * 2 + 1] = rsqrtf(var + 1e-5f);
    }
    __syncthreads();
    {
        const int e = tid >> 3, seg = tid & 7;
        const float mu = lnmv[e * 2], rs = lnmv[e * 2 + 1];
        for (int k = seg * 16; k < seg * 16 + 16; ++k) {
            float v = (hbuf[e * XS + k] - mu) * rs * g1[k] + be1[k];
            v = v / (1.0f + __expf(-v));         // SiLU
            hbuf[e * XS + k] = v;
        }
    }
    __syncthreads();

    // ---- GEMM2: rad[16x896] = h @ W2 + b2 ----
    for (int nt = wave; nt < OUTD / 16; nt += 4) {
        const int n0 = nt * 16;
        v8f acc = {};
        for (int k0 = 0; k0 < HID; k0 += 4) {
            const int ka = k0 + half * 2;
            v2f a = *(const v2f*)(hbuf + lm * XS + ka);       // ds_load_b64
            v2f b;
            b[0] = W2[(size_t)ka * OUTD + n0 + lm];
            b[1] = W2[(size_t)(ka + 1) * OUTD + n0 + lm];
            acc = __builtin_amdgcn_wmma_f32_16x16x4_f32(false, a, false, b,
                                                        (short)0, acc, false, false);
        }
        const float bb = b2[n0 + lm];
        #pragma unroll
        for (int i = 0; i < 8; ++i)
            radbuf[(i + half * 8) * RADS + n0 + lm] = acc[i] + bb;
    }
    __syncthreads();

    // ---- einsum emb[e,f,c] = sum_m wig[e,f,m]*rad[e,m,c], scatter-add ----
    const int ch = tid;   // channel 0..127
    for (int e = 0; e < nE; ++e) {
        const float* wr = wbuf + e * FULLC * M0;
        const float* rr = radbuf + e * RADS;
        const int dst = eidx[E + edge0 + e];            // edge_index[1][edge]
        float* ob = out + (size_t)dst * (FULLC * CCH) + ch;
        const float r0 = rr[0 * CCH + ch], r1 = rr[1 * CCH + ch],
                    r2 = rr[2 * CCH + ch], r3 = rr[3 * CCH + ch],
                    r4 = rr[4 * CCH + ch], r5 = rr[5 * CCH + ch],
                    r6 = rr[6 * CCH + ch];
        for (int f = 0; f < FULLC; ++f) {
            const float* w7 = wr + f * M0;              // LDS broadcast reads
            float a = w7[0] * r0 + w7[1] * r1 + w7[2] * r2 + w7[3] * r3
                    + w7[4] * r4 + w7[5] * r5 + w7[6] * r6;
            atom_add_f32(ob + f * CCH, a * INV_RESCALE);
        }
    }
}

extern "C" void kernel_launch(void* const* d_in, const int* in_sizes, int n_in,
                              void* d_out, int out_size, void* d_ws, size_t ws_size,
                              hipStream_t stream) {
    const float* x      = (const float*)d_in[0];
    const float* x_edge = (const float*)d_in[1];
    const float* wig    = (const float*)d_in[2];
    const float* W1     = (const float*)d_in[3];
    const float* b1     = (const float*)d_in[4];
    const float* g1     = (const float*)d_in[5];
    const float* be1    = (const float*)d_in[6];
    const float* W2     = (const float*)d_in[7];
    const float* b2     = (const float*)d_in[8];
    const int*   eidx   = (const int*)d_in[9];
    float* out = (float*)d_out;

    const int E = in_sizes[1] / HID;

    // out = x
    const int n4 = out_size / 4;
    copy_f4_kernel<<<(n4 + 255) / 256, 256, 0, stream>>>(
        (const float4*)x, (float4*)out, n4);

    // fused MLP -> wigner einsum -> scatter-add
    const int nblocks = (E + TE - 1) / TE;
    const size_t ldsBytes = (size_t)LDS_FLOATS * sizeof(float);
    edge_deg_fused_kernel<<<nblocks, 128, ldsBytes, stream>>>(
        x_edge, wig, W1, b1, g1, be1, W2, b2, eidx, out, E);
}